// BlockSparseMatrix_27401891349167
// MI455X (gfx1250) — compile-verified
//
#include <hip/hip_runtime.h>

// Problem constants (fixed by the reference: SHAPE=(8192,8192), BH=BW=32)
#define XDIM   256      // block-rows
#define YDIM   256      // block-cols
#define BH     32
#define BW     32
#define WIDTH  8192     // dense row length
#define NPOS   (XDIM * YDIM)   // 65536 mask entries
#define LSTR   36       // padded LDS row stride in floats (16B-aligned, low bank conflict)

typedef float v4f __attribute__((ext_vector_type(4)));
typedef int   v4i __attribute__((ext_vector_type(4)));

// Address-space-qualified pointer types for the async builtin:
// diagnostic showed param0 = "int __vector(4) __device__ *"  (AS1 global)
typedef __attribute__((address_space(1))) v4i gv4i;   // global int4
typedef __attribute__((address_space(3))) v4i lv4i;   // LDS int4

// ---- CDNA5 async global->LDS path (probe via __has_builtin; fallback = sync load) ----
#if defined(__has_builtin)
#  if __has_builtin(__builtin_amdgcn_global_load_async_to_lds_b128)
#    define HAS_ASYNC_LDS 1
#  endif
#endif
#ifndef HAS_ASYNC_LDS
#  define HAS_ASYNC_LDS 0
#endif

// =====================================================================
// Kernel 1: build all index arrays + rank table (single workgroup).
// mask is numpy/jax bool -> 1 byte per element.
// Integer outputs are written as f32 (d_out is a flat f32 buffer).
// =====================================================================
__global__ __launch_bounds__(1024) void bsm_build_index_kernel(
    const unsigned char* __restrict__ mask,
    float* __restrict__ blocks,    // [2n] interleaved [col,row]
    float* __restrict__ cols_a,    // [2n] interleaved [col, block_ptr]
    float* __restrict__ rse_a,     // [XDIM+1] CSR row pointers
    float* __restrict__ rows_b,    // [2n] interleaved [row, block_ptr_t]
    float* __restrict__ cse_b,     // [YDIM+1] CSC col pointers
    int*   __restrict__ rank_ws)   // [NPOS] row-major rank per position (only set where mask=1)
{
    __shared__ int sc[1025];
    const int tid = threadIdx.x;
    const int per = NPOS / 1024;          // 64 positions per thread
    const int base_p = tid * per;

    // ---------------- pass 1: row-major scan ----------------
    int cnt = 0;
    for (int q = 0; q < per; ++q) cnt += mask[base_p + q] ? 1 : 0;
    sc[tid] = cnt;
    __syncthreads();
    if (tid == 0) {
        int run = 0;
        for (int i = 0; i < 1024; ++i) { int t = sc[i]; sc[i] = run; run += t; }
        sc[1024] = run;
    }
    __syncthreads();
    int rank  = sc[tid];
    const int total = sc[1024];
    __syncthreads();   // everyone has read sc; safe to reuse below

    for (int q = 0; q < per; ++q) {
        const int p = base_p + q;
        if ((p & (YDIM - 1)) == 0) rse_a[p >> 8] = (float)rank;   // exclusive count at row start
        if (mask[p]) {
            const int r = p >> 8, c = p & (YDIM - 1);
            blocks[2 * rank + 0] = (float)c;
            blocks[2 * rank + 1] = (float)r;
            cols_a[2 * rank + 0] = (float)c;
            cols_a[2 * rank + 1] = (float)rank;   // block_ptr = identity
            rank_ws[p] = rank;
            ++rank;
        }
    }
    if (tid == 0) rse_a[XDIM] = (float)total;

    // ---------------- pass 2: col-major scan (transposed mask) ----------------
    cnt = 0;
    for (int q = 0; q < per; ++q) {
        const int qq = base_p + q;
        const int c = qq >> 8, r = qq & (XDIM - 1);   // maskT[c][r]
        cnt += mask[r * YDIM + c] ? 1 : 0;
    }
    sc[tid] = cnt;
    __syncthreads();
    if (tid == 0) {
        int run = 0;
        for (int i = 0; i < 1024; ++i) { int t = sc[i]; sc[i] = run; run += t; }
        sc[1024] = run;
    }
    __syncthreads();
    int trank = sc[tid];

    for (int q = 0; q < per; ++q) {
        const int qq = base_p + q;
        const int c = qq >> 8, r = qq & (XDIM - 1);
        if ((qq & (XDIM - 1)) == 0) cse_b[c] = (float)trank;      // exclusive count at col start
        const int p = r * YDIM + c;
        if (mask[p]) {
            rows_b[2 * trank + 0] = (float)r;                     // cols_t = original row
            rows_b[2 * trank + 1] = (float)rank_ws[p];            // block_ptr_t = row-major rank
            ++trank;
        }
    }
    if (tid == 0) cse_b[YDIM] = (float)total;
}

// =====================================================================
// Kernel 2: one workgroup per 32x32 tile. Empty tile -> NT zero store.
// Occupied tile -> async global->LDS copy, transpose in LDS, NT store.
// Writes every dense element exactly once.
// =====================================================================
__global__ __launch_bounds__(256) void bsm_scatter_kernel(
    const unsigned char* __restrict__ mask,
    const float* __restrict__ data,
    const int*   __restrict__ rank_ws,
    float*       __restrict__ dense)
{
    const int p   = blockIdx.x;            // tile id (row-major over 256x256)
    const int r   = p >> 8;
    const int c   = p & (YDIM - 1);
    const int tid = threadIdx.x;
    const int i   = tid >> 3;              // dense row within tile  (0..31)
    const int k   = tid & 7;               // float4 chunk           (0..7)

    float* drow = dense + (size_t)(r * BH + i) * WIDTH + (size_t)c * BW + k * 4;

    if (!mask[p]) {                        // empty tile: stream zeros
        v4f z = {0.f, 0.f, 0.f, 0.f};
        __builtin_nontemporal_store(z, (v4f*)drow);
        return;                            // uniform per block: no barrier divergence
    }

    __shared__ float lds[BH * LSTR];       // padded: byte stride 144 (16B aligned), stride-36 reads

    const int    n   = rank_ws[p];
    const float* src = data + (size_t)n * (BH * BW) + (size_t)tid * 4;   // row j=i, col k*4
    float*       ldp = &lds[i * LSTR + k * 4];                           // 16B-aligned LDS slot

#if HAS_ASYNC_LDS
    __builtin_amdgcn_global_load_async_to_lds_b128((gv4i*)src, (lv4i*)ldp,
                                                   /*offset=*/0, /*cpol=*/0);
#  if __has_builtin(__builtin_amdgcn_s_wait_asynccnt)
    __builtin_amdgcn_s_wait_asynccnt(0);
#  else
    asm volatile("s_wait_asynccnt 0" ::: "memory");
#  endif
#else
    v4f v = __builtin_nontemporal_load((const v4f*)src);
    *(v4f*)ldp = v;
#endif
    __syncthreads();

    // dense[r*32+i, c*32+jj] = data[n, jj, i]  (block stored transposed)
    const int jj0 = k * 4;
    v4f o;
    o.x = lds[(jj0 + 0) * LSTR + i];
    o.y = lds[(jj0 + 1) * LSTR + i];
    o.z = lds[(jj0 + 2) * LSTR + i];
    o.w = lds[(jj0 + 3) * LSTR + i];
    __builtin_nontemporal_store(o, (v4f*)drow);
}

// =====================================================================
// Host launcher
// =====================================================================
extern "C" void kernel_launch(void* const* d_in, const int* in_sizes, int n_in,
                              void* d_out, int out_size, void* d_ws, size_t ws_size,
                              hipStream_t stream) {
    const unsigned char* mask = (const unsigned char*)d_in[0];  // jnp bool -> 1 byte/elem
    const float*         data = (const float*)d_in[1];
    const int n = in_sizes[1] / (BH * BW);                      // 16384 blocks

    float* out    = (float*)d_out;
    float* dense  = out;
    float* blocks = dense  + (size_t)WIDTH * WIDTH;   // 8192*8192
    float* cols_a = blocks + (size_t)2 * n;
    float* rse_a  = cols_a + (size_t)2 * n;
    float* rows_b = rse_a  + (XDIM + 1);
    float* cse_b  = rows_b + (size_t)2 * n;

    int* rank_ws = (int*)d_ws;                        // 65536 ints = 256 KB scratch

    bsm_build_index_kernel<<<1, 1024, 0, stream>>>(mask, blocks, cols_a, rse_a,
                                                   rows_b, cse_b, rank_ws);
    bsm_scatter_kernel<<<NPOS, 256, 0, stream>>>(mask, data, rank_ws, dense);
}